// HermitePolynomialMatrixND_13030930776101
// MI455X (gfx1250) — compile-verified
//
#include <hip/hip_runtime.h>
#include <hip/hip_bf16.h>
#include <stdint.h>

// HermitePolynomialMatrixND: out[w,0,a,p] = env(w,a) * N(q0)N(q1)N(q2)
//                                          * H_q0(x0) H_q1(x1) H_q2(x2)
// Multi-index table (stable sort of 3-tuples by sum, first 16):
//  0:(0,0,0) 1:(0,0,1) 2:(0,1,0) 3:(1,0,0) 4:(0,0,2) 5:(0,1,1) 6:(0,2,0)
//  7:(1,0,1) 8:(1,1,0) 9:(2,0,0) 10:(0,0,3) 11:(0,1,2) 12:(0,2,1) 13:(0,3,0)
// 14:(1,0,2) 15:(1,1,1)   => max order 3, q0 <= 2.
// Roofline: 24 MiB in + 128 MiB out = 152 MiB @ 23.3 TB/s  -> ~6.5 us floor;
// ~60 flops per (w,a) -> hopelessly memory-bound; no matrix contraction exists
// (triple-gather *product*), so WMMA is inapplicable. Optimize data movement:
// async global->LDS in, async LDS->global out (dense 512B per instruction).

#define WPB 16    // walkers per block
#define TPB 256   // 16 walkers * 16 particles

// N(n) = pi^{-1/4} / sqrt(2^n n!)
#define NH0 0.75112554446494248f
#define NH1 0.53112596601359856f
#define NH2 0.26556298300679928f
#define NH3 0.10841563400000000f  // pi^-1/4 / sqrt(48)

__global__ __launch_bounds__(TPB) void hermite_orb_kernel(
    const float* __restrict__ x, float* __restrict__ out, int nwalk) {
  __shared__ float sx[WPB * 48];   // 3 KiB input tile (contiguous in global)
  __shared__ float so[TPB * 16];   // 16 KiB output tile, exact linear image

  const int t   = threadIdx.x;
  const int blk = blockIdx.x;
  const long long wbase = (long long)blk * WPB;

  int wcnt = nwalk - (int)wbase;   // block-uniform tail handling
  if (wcnt > WPB) wcnt = WPB;
  if (wcnt <= 0) return;

  const float* gsrc = x + wbase * 48;    // 48 floats / walker
  float*       gdst = out + wbase * 256; // 256 floats / walker (16a * 16p)

#if defined(__gfx1250__)
  // ---- async global -> LDS staging (ASYNCcnt, no VGPR landing) ----
  {
    const int n16 = (wcnt * 192) / 16;   // 192 B per walker, b128 chunks
    if (t < n16) {
      unsigned goff = (unsigned)t * 16u;
      // low 32 bits of a flat shared-space pointer == logical LDS address
      unsigned ldsa = (unsigned)(uintptr_t)(&sx[0]) + goff;
      asm volatile("global_load_async_to_lds_b128 %0, %1, %2"
                   :
                   : "v"(ldsa), "v"(goff), "s"(gsrc)
                   : "memory");
    }
    asm volatile("s_wait_asynccnt 0" ::: "memory");
  }
#else
  for (int i = t; i < wcnt * 48; i += TPB) sx[i] = gsrc[i];
#endif
  __syncthreads();

  // ---- compute one (walker, particle) per thread ----
  // Lanes past the tail compute on stale LDS; their results are never drained.
  const float* sp = &sx[t * 3];
  const float x0 = sp[0], x1 = sp[1], x2 = sp[2];

  const float env = __expf(-0.5f * (x0 * x0 + x1 * x1 + x2 * x2));

  // dim 0 (x): orders 0..2, envelope folded in
  const float u0 = 2.0f * x0;
  const float a0 = env * NH0;
  const float a1 = env * (NH1 * u0);
  const float a2 = env * (NH2 * (u0 * u0 - 2.0f));  // H2 = 4x^2 - 2

  // dim 1 (y): orders 0..3  (H3 = u*(H2-4), u = 2y)
  const float u1  = 2.0f * x1;
  const float H2y = u1 * u1 - 2.0f;
  const float b1  = NH1 * u1;
  const float b2  = NH2 * H2y;
  const float b3  = NH3 * (u1 * (H2y - 4.0f));

  // dim 2 (z): orders 0..3
  const float u2  = 2.0f * x2;
  const float H2z = u2 * u2 - 2.0f;
  const float c1  = NH1 * u2;
  const float c2  = NH2 * H2z;
  const float c3  = NH3 * (u2 * (H2z - 4.0f));

  // shared x*y products (b0 = c0 = NH0)
  const float ab00 = a0 * NH0;
  const float ab01 = a0 * b1;
  const float ab02 = a0 * b2;
  const float ab03 = a0 * b3;
  const float ab10 = a1 * NH0;
  const float ab11 = a1 * b1;
  const float ab20 = a2 * NH0;

  float4 o0, o1, o2, o3;
  o0.x = ab00 * NH0;  // p0  (0,0,0)
  o0.y = ab00 * c1;   // p1  (0,0,1)
  o0.z = ab01 * NH0;  // p2  (0,1,0)
  o0.w = ab10 * NH0;  // p3  (1,0,0)
  o1.x = ab00 * c2;   // p4  (0,0,2)
  o1.y = ab01 * c1;   // p5  (0,1,1)
  o1.z = ab02 * NH0;  // p6  (0,2,0)
  o1.w = ab10 * c1;   // p7  (1,0,1)
  o2.x = ab11 * NH0;  // p8  (1,1,0)
  o2.y = ab20 * NH0;  // p9  (2,0,0)
  o2.z = ab00 * c3;   // p10 (0,0,3)
  o2.w = ab01 * c2;   // p11 (0,1,2)
  o3.x = ab02 * c1;   // p12 (0,2,1)
  o3.y = ab03 * NH0;  // p13 (0,3,0)
  o3.z = ab10 * c2;   // p14 (1,0,2)
  o3.w = ab11 * c1;   // p15 (1,1,1)

  // park the 16 outputs in the LDS linear image of the block's output tile
  float4* lp = reinterpret_cast<float4*>(&so[t * 16]);
  lp[0] = o0;
  lp[1] = o1;
  lp[2] = o2;
  lp[3] = o3;
  __syncthreads();

#if defined(__gfx1250__)
  // ---- async LDS -> global drain: each instruction moves a dense 512 B ----
  {
    const int nchunk = wcnt * 64;  // valid float4 chunks in the tile
    const unsigned lbase = (unsigned)(uintptr_t)(&so[0]);
#pragma unroll
    for (int i = 0; i < 4; ++i) {
      const int c = i * TPB + t;
      if (c < nchunk) {
        unsigned goff = (unsigned)c * 16u;
        unsigned ldsa = lbase + goff;
        asm volatile("global_store_async_from_lds_b128 %0, %1, %2 th:TH_STORE_NT"
                     :
                     : "v"(goff), "v"(ldsa), "s"(gdst)
                     : "memory");
      }
    }
    // S_ENDPGM performs an implicit wait-idle, covering outstanding ASYNCcnt.
  }
#else
  for (int i = 0; i < 4; ++i) {
    const int c = i * TPB + t;
    if (c < wcnt * 64)
      reinterpret_cast<float4*>(gdst)[c] = reinterpret_cast<const float4*>(so)[c];
  }
#endif
}

extern "C" void kernel_launch(void* const* d_in, const int* in_sizes, int n_in,
                              void* d_out, int out_size, void* d_ws, size_t ws_size,
                              hipStream_t stream) {
  const float* x = (const float*)d_in[0];
  float* out = (float*)d_out;
  const int nwalk = in_sizes[0] / 48;          // x is [W,16,3] f32
  const int blocks = (nwalk + WPB - 1) / WPB;  // 8192 for W=131072
  hermite_orb_kernel<<<blocks, TPB, 0, stream>>>(x, out, nwalk);
}